// TokenTreeModel_32873679684166
// MI455X (gfx1250) — compile-verified
//
#include <hip/hip_runtime.h>

// TokenTreeModel scatter + depth-linear, MI455X (gfx1250, wave32).
//
// out[b,t,v] = bias + sum_{d,c} (child_tokens[b,t,d,c]==v) * child_counts[b,t,d,c] * w[d]
//
// Strategy: one workgroup per (b,t). Keep the entire 32000-float vocab row in
// LDS (128 KB -- CDNA5 WGP has 320 KB LDS), stage the 4 KB of per-(b,t) node
// data with gfx1250 async global->LDS DMA (ASYNCcnt), scatter with ds_add_f32,
// then stream the row out with 128-bit stores. Bandwidth-bound: ~65.5 MB of
// output => ~3 us floor at 23.3 TB/s.

#define BDIM   2
#define TDIM   256
#define DEPTH  8
#define VOCAB  32000
#define NCHILD 64
#define NODES  (DEPTH * NCHILD)   // 512 nodes per (b,t)
#define BLOCK  512                // 16 wave32s

__global__ __launch_bounds__(BLOCK, 1)
void token_tree_scatter_kernel(const int* __restrict__ child_tokens,
                               const float* __restrict__ child_counts,
                               const float* __restrict__ w,
                               const float* __restrict__ bias,
                               float* __restrict__ out)
{
    __shared__ float hist[VOCAB];     // 128000 B vocab accumulator
    __shared__ int   s_tok[NODES];    //   2048 B
    __shared__ float s_cnt[NODES];    //   2048 B

    const int bt  = blockIdx.x;       // 0 .. B*T-1
    const int tid = threadIdx.x;      // 0 .. 511

    // ---- Stage this (b,t)'s tokens + counts into LDS via async DMA.
    // 512 ints = 2048 B = 128 lanes x b128. Lanes 0..127 pull tokens,
    // lanes 128..255 pull counts. Tracked by ASYNCcnt (gfx1250 path).
    if (tid < 128) {
        const int* g = child_tokens + (size_t)bt * NODES + (size_t)tid * 4;
        unsigned lds = (unsigned)(size_t)&s_tok[tid * 4];   // low 32 bits = LDS offset
        asm volatile("global_load_async_to_lds_b128 %0, %1, off"
                     :: "v"(lds), "v"(g) : "memory");
    } else if (tid < 256) {
        const int i = tid - 128;
        const float* g = child_counts + (size_t)bt * NODES + (size_t)i * 4;
        unsigned lds = (unsigned)(size_t)&s_cnt[i * 4];
        asm volatile("global_load_async_to_lds_b128 %0, %1, off"
                     :: "v"(lds), "v"(g) : "memory");
    }

    // ---- Zero the vocab histogram while the DMA is in flight.
    #pragma unroll 4
    for (int i = tid; i < VOCAB; i += BLOCK) hist[i] = 0.0f;

    asm volatile("s_wait_asynccnt 0" ::: "memory");
    __syncthreads();

    // ---- Scatter: one (depth, child) node per thread; fold w[d] into the
    // count so the depth-8 "Linear" never needs a materialized [depth,vocab].
    {
        const int   d       = tid >> 6;          // tid / NCHILD
        const int   tok     = s_tok[tid];
        const float contrib = s_cnt[tid] * w[d];
        atomicAdd(&hist[tok], contrib);          // ds_add_f32 (no return)
    }
    __syncthreads();

    // ---- Stream out bias + hist with 128-bit loads/stores (coalesced).
    const float bv = bias[0];
    float4* __restrict__ out4 = (float4*)(out + (size_t)bt * VOCAB);
    const float4* h4 = (const float4*)hist;
    #pragma unroll 4
    for (int i = tid; i < VOCAB / 4; i += BLOCK) {   // 8000 float4s per row
        float4 v = h4[i];
        v.x += bv; v.y += bv; v.z += bv; v.w += bv;
        out4[i] = v;
    }
}

extern "C" void kernel_launch(void* const* d_in, const int* in_sizes, int n_in,
                              void* d_out, int out_size, void* d_ws, size_t ws_size,
                              hipStream_t stream) {
    const int*   child_tokens = (const int*)  d_in[0];
    const float* child_counts = (const float*)d_in[1];
    const float* w            = (const float*)d_in[2];
    const float* bias         = (const float*)d_in[3];
    float* out = (float*)d_out;

    dim3 grid(BDIM * TDIM);   // 512 workgroups, one per (b,t)
    dim3 block(BLOCK);        // 512 threads = 16 wave32s
    token_tree_scatter_kernel<<<grid, block, 0, stream>>>(
        child_tokens, child_counts, w, bias, out);
}